// SAMBA_89730456748198
// MI455X (gfx1250) — compile-verified
//
#include <hip/hip_runtime.h>
#include <math.h>

typedef __attribute__((ext_vector_type(2))) float v2f;
typedef __attribute__((ext_vector_type(8))) float v8f;

#define N_NODES 2048
#define EMBED   32
#define BATCH   16
#define INP     32
#define OUTC    32
#define BC      512   /* BATCH*INP: stacked column count for A@X */
#define KIO     3072  /* CHEB_K*INP*OUTC */

// ---------------------------------------------------------------------------
// Transpose x: input viewed as (512, 2048) row-major -> X (2048, 512)
// ---------------------------------------------------------------------------
__global__ void samba_transpose_x(const float* __restrict__ in, float* __restrict__ out) {
    __shared__ float tile[32][33];
    int tx = threadIdx.x & 31;
    int ty = threadIdx.x >> 5;          // 0..7
    int bx = blockIdx.x;                // col tile over 2048/32 = 64
    int by = blockIdx.y;                // row tile over 512/32  = 16
#pragma unroll
    for (int rr = 0; rr < 4; ++rr) {
        int r = by * 32 + ty + rr * 8;
        tile[ty + rr * 8][tx] = in[r * N_NODES + bx * 32 + tx];
    }
    __syncthreads();
#pragma unroll
    for (int rr = 0; rr < 4; ++rr) {
        int r = bx * 32 + ty + rr * 8;  // output row (node n)
        out[r * BC + by * 32 + tx] = tile[tx][ty + rr * 8];
    }
}

// ---------------------------------------------------------------------------
// A = softmax_row(exp(-gamma * ||a_i - a_j||^2)) * drop_mask
// One block (256 threads) per row i; each thread owns 8 columns.
// ---------------------------------------------------------------------------
__global__ void samba_adjacency(const float* __restrict__ adj, const float* __restrict__ gamma,
                                const float* __restrict__ drop, float* __restrict__ A) {
    __shared__ float ai[EMBED];
    __shared__ float red[256];
    const int tid = threadIdx.x;
    const int row = blockIdx.x;
    if (tid < EMBED) ai[tid] = adj[row * EMBED + tid];
    __syncthreads();
    const float g = gamma[0];

    float z[8];
    float mx = -INFINITY;
#pragma unroll
    for (int jj = 0; jj < 8; ++jj) {
        int j = jj * 256 + tid;
        const float4* aj4 = (const float4*)(adj + j * EMBED);
        float d = 0.0f;
#pragma unroll
        for (int e = 0; e < 8; ++e) {
            float4 v = aj4[e];
            float t0 = ai[e * 4 + 0] - v.x;
            float t1 = ai[e * 4 + 1] - v.y;
            float t2 = ai[e * 4 + 2] - v.z;
            float t3 = ai[e * 4 + 3] - v.w;
            d += t0 * t0 + t1 * t1 + t2 * t2 + t3 * t3;
        }
        float zz = __expf(-g * d);
        z[jj] = zz;
        mx = fmaxf(mx, zz);
    }
    // block max
    red[tid] = mx;
    __syncthreads();
    for (int s = 128; s > 0; s >>= 1) {
        if (tid < s) red[tid] = fmaxf(red[tid], red[tid + s]);
        __syncthreads();
    }
    mx = red[0];
    __syncthreads();
    // exp + block sum
    float sum = 0.0f;
#pragma unroll
    for (int jj = 0; jj < 8; ++jj) {
        float e = __expf(z[jj] - mx);
        z[jj] = e;
        sum += e;
    }
    red[tid] = sum;
    __syncthreads();
    for (int s = 128; s > 0; s >>= 1) {
        if (tid < s) red[tid] += red[tid + s];
        __syncthreads();
    }
    const float inv = 1.0f / red[0];
#pragma unroll
    for (int jj = 0; jj < 8; ++jj) {
        int j = jj * 256 + tid;
        A[row * N_NODES + j] = z[jj] * inv * drop[row * N_NODES + j];
    }
}

// ---------------------------------------------------------------------------
// fp32 WMMA GEMM, 32x32 tile per wave (4 accumulators):
//   D = alpha * A(MxK) @ B(KxN) + beta * Csrc(MxN)
// NCOLS is a compile-time template arg so B-operand row offsets become
// immediate load offsets and the K-loop is pure pointer-bump.
// Requires: M % 32 == 0, K % 4 == 0, NCOLS % 32 == 0.
// ---------------------------------------------------------------------------
template <int NCOLS>
__global__ void samba_gemm32(const float* __restrict__ A, const float* __restrict__ B,
                             const float* __restrict__ Csrc, float* __restrict__ D,
                             int M, int Kd, float alpha, float beta) {
    const int lane   = threadIdx.x & 31;
    const int waveId = threadIdx.x >> 5;
    const int colTiles = NCOLS >> 5;          // 32-wide col tiles
    const int rowTiles = M >> 5;              // 32-tall row tiles
    const int tileId = blockIdx.x * (blockDim.x >> 5) + waveId;
    if (tileId >= rowTiles * colTiles) return;
    const int rowT = tileId / colTiles;
    const int colT = tileId - rowT * colTiles;

    const int m16 = lane & 15;                // A row / B col within 16-tile
    const int klo = (lane >> 4) * 2;          // lanes 0-15: K 0,1 ; 16-31: K 2,3

    const float* a0 = A + (size_t)(rowT * 32 + m16) * Kd + klo;       // rows 0..15
    const float* a1 = A + (size_t)(rowT * 32 + 16 + m16) * Kd + klo;  // rows 16..31
    const float* bp = B + (size_t)klo * NCOLS + colT * 32 + m16;

    v8f acc00 = {}, acc01 = {}, acc10 = {}, acc11 = {};
#pragma unroll 4
    for (int m = 0; m < Kd; m += 4) {
        v2f aA = *(const v2f*)a0;
        v2f aB = *(const v2f*)a1;
        v2f b0, b1;
        b0.x = bp[0];
        b0.y = bp[NCOLS];
        b1.x = bp[16];
        b1.y = bp[NCOLS + 16];
        acc00 = __builtin_amdgcn_wmma_f32_16x16x4_f32(false, aA, false, b0, (short)0, acc00, false, false);
        acc01 = __builtin_amdgcn_wmma_f32_16x16x4_f32(false, aA, false, b1, (short)0, acc01, false, false);
        acc10 = __builtin_amdgcn_wmma_f32_16x16x4_f32(false, aB, false, b0, (short)0, acc10, false, false);
        acc11 = __builtin_amdgcn_wmma_f32_16x16x4_f32(false, aB, false, b1, (short)0, acc11, false, false);
        a0 += 4;
        a1 += 4;
        bp += 4 * NCOLS;
    }

    const int half = lane >> 4;
    const int baseRow = rowT * 32 + 8 * half;
    const int baseCol = colT * 32 + m16;
#pragma unroll
    for (int r = 0; r < 8; ++r) {
        size_t i00 = (size_t)(baseRow + r) * NCOLS + baseCol;           // rows 0..15,  cols 0..15
        size_t i01 = i00 + 16;                                          // cols 16..31
        size_t i10 = (size_t)(baseRow + 16 + r) * NCOLS + baseCol;      // rows 16..31
        size_t i11 = i10 + 16;
        float v00 = alpha * acc00[r];
        float v01 = alpha * acc01[r];
        float v10 = alpha * acc10[r];
        float v11 = alpha * acc11[r];
        if (beta != 0.0f) {
            v00 += beta * Csrc[i00];
            v01 += beta * Csrc[i01];
            v10 += beta * Csrc[i10];
            v11 += beta * Csrc[i11];
        }
        D[i00] = v00;
        D[i01] = v01;
        D[i10] = v10;
        D[i11] = v11;
    }
}

// ---------------------------------------------------------------------------
// Per-node dynamic graph-conv: one wave per node.
//   out[b,n,o] = sum_{kk<96} xg[n][b][kk] * W[n][kk][o] + bias[n][o]
// xg rows come from {X, Y1, Y2} (kk blocks of 32), M=16 (batch), N=32 (2 tiles).
// ---------------------------------------------------------------------------
__global__ void samba_node_out(const float* __restrict__ Xmat, const float* __restrict__ Y1,
                               const float* __restrict__ Y2, const float* __restrict__ W,
                               const float* __restrict__ biasMat, float* __restrict__ outFull) {
    const int lane   = threadIdx.x & 31;
    const int waveId = threadIdx.x >> 5;
    const int node   = blockIdx.x * (blockDim.x >> 5) + waveId;

    const int m16 = lane & 15;          // batch row for A-frag / out col for B-frag
    const int klo = (lane >> 4) * 2;

    const float* Wrow = W + (size_t)node * KIO + klo * OUTC + m16;
    v8f acc0 = {};
    v8f acc1 = {};
#pragma unroll
    for (int s = 0; s < 3; ++s) {
        const float* src = (s == 0) ? Xmat : (s == 1 ? Y1 : Y2);
        const float* P = src + (size_t)node * BC + m16 * INP + klo;
#pragma unroll
        for (int kk2 = 0; kk2 < 32; kk2 += 4) {
            v2f a = *(const v2f*)(P + kk2);
            const int wbase = (s * 32 + kk2) * OUTC;
            v2f b0, b1;
            b0.x = Wrow[wbase];
            b0.y = Wrow[wbase + OUTC];
            b1.x = Wrow[wbase + 16];
            b1.y = Wrow[wbase + OUTC + 16];
            acc0 = __builtin_amdgcn_wmma_f32_16x16x4_f32(false, a, false, b0, (short)0, acc0, false, false);
            acc1 = __builtin_amdgcn_wmma_f32_16x16x4_f32(false, a, false, b1, (short)0, acc1, false, false);
        }
    }
    const int half = lane >> 4;
    const float* brow = biasMat + (size_t)node * OUTC;
#pragma unroll
    for (int r = 0; r < 8; ++r) {
        int b  = r + 8 * half;                 // batch index
        int o0 = m16;
        size_t base = ((size_t)b * N_NODES + node) * OUTC;
        outFull[base + o0]      = acc0[r] + brow[o0];
        outFull[base + o0 + 16] = acc1[r] + brow[o0 + 16];
    }
}

// ---------------------------------------------------------------------------
// Projection over nodes: out[b,o] = sum_n outFull[b,n,o] * pw[n] + pb
// One block per batch b; 256 threads = 32 o-lanes x 8 n-chunks.
// ---------------------------------------------------------------------------
__global__ void samba_proj(const float* __restrict__ outFull, const float* __restrict__ pw,
                           const float* __restrict__ pb, float* __restrict__ out) {
    __shared__ float sh[256];
    const int tid = threadIdx.x;
    const int b = blockIdx.x;
    const int o = tid & 31;
    const int chunk = tid >> 5;            // 0..7
    float partial = 0.0f;
    for (int n = chunk * 256; n < chunk * 256 + 256; ++n) {
        partial += outFull[((size_t)b * N_NODES + n) * OUTC + o] * pw[n];
    }
    sh[tid] = partial;
    __syncthreads();
    if (tid < 32) {
        float s = 0.0f;
#pragma unroll
        for (int c = 0; c < 8; ++c) s += sh[c * 32 + tid];
        out[b * OUTC + tid] = s + pb[0];
    }
}

// ---------------------------------------------------------------------------
extern "C" void kernel_launch(void* const* d_in, const int* in_sizes, int n_in,
                              void* d_out, int out_size, void* d_ws, size_t ws_size,
                              hipStream_t stream) {
    (void)in_sizes; (void)n_in; (void)out_size; (void)ws_size;
    const float* x      = (const float*)d_in[0];  // (16,32,2048)
    const float* adj    = (const float*)d_in[1];  // (2048,32)
    const float* gamma  = (const float*)d_in[2];  // scalar
    const float* wpool  = (const float*)d_in[3];  // (32,3,32,32) == (32,3072)
    const float* bpool  = (const float*)d_in[4];  // (32,32)
    const float* projw  = (const float*)d_in[5];  // (1,2048)
    const float* projb  = (const float*)d_in[6];  // (1,)
    const float* dropm  = (const float*)d_in[7];  // (2048,2048)
    float* out = (float*)d_out;

    float* ws = (float*)d_ws;
    float* A_mat   = ws;                         // 2048*2048
    float* X_mat   = A_mat   + 2048u * 2048u;    // 2048*512
    float* Y1      = X_mat   + 2048u * 512u;     // 2048*512
    float* Y2      = Y1      + 2048u * 512u;     // 2048*512
    float* W_mat   = Y2      + 2048u * 512u;     // 2048*3072
    float* B_mat   = W_mat   + 2048u * 3072u;    // 2048*32
    float* outFull = B_mat   + 2048u * 32u;      // 16*2048*32

    // 1) transpose x -> X (2048 x 512)
    samba_transpose_x<<<dim3(64, 16), 256, 0, stream>>>(x, X_mat);

    // 2) adjacency (dist -> exp -> softmax -> dropout)
    samba_adjacency<<<2048, 256, 0, stream>>>(adj, gamma, dropm, A_mat);

    // 3) W_mat = adj @ wpool   (2048 x 32 x 3072)
    {
        int tiles = (2048 / 32) * (KIO / 32);
        samba_gemm32<KIO><<<(tiles + 7) / 8, 256, 0, stream>>>(
            adj, wpool, nullptr, W_mat, 2048, EMBED, 1.0f, 0.0f);
    }
    // 4) B_mat = adj @ bpool   (2048 x 32 x 32)
    {
        int tiles = (2048 / 32) * (OUTC / 32);
        samba_gemm32<OUTC><<<(tiles + 7) / 8, 256, 0, stream>>>(
            adj, bpool, nullptr, B_mat, 2048, EMBED, 1.0f, 0.0f);
    }
    // 5) Y1 = A @ X            (2048 x 2048 x 512)
    {
        int tiles = (2048 / 32) * (BC / 32);
        samba_gemm32<BC><<<(tiles + 7) / 8, 256, 0, stream>>>(
            A_mat, X_mat, nullptr, Y1, 2048, 2048, 1.0f, 0.0f);
    }
    // 6) Y2 = 2*(A @ Y1) - X   (Chebyshev T2 applied to X without forming T2)
    {
        int tiles = (2048 / 32) * (BC / 32);
        samba_gemm32<BC><<<(tiles + 7) / 8, 256, 0, stream>>>(
            A_mat, Y1, X_mat, Y2, 2048, 2048, 2.0f, -1.0f);
    }
    // 7) per-node (16x96)@(96x32) + bias -> outFull
    samba_node_out<<<2048 / 8, 256, 0, stream>>>(X_mat, Y1, Y2, W_mat, B_mat, outFull);

    // 8) projection over nodes -> (16,32,1)
    samba_proj<<<16, 256, 0, stream>>>(outFull, projw, projb, out);
}